// SCGCN_39994735460984
// MI455X (gfx1250) — compile-verified
//
#include <hip/hip_runtime.h>
#include <cstdint>
#include <cstddef>

// ---------------------------------------------------------------------------
// SCGCN on MI455X (gfx1250, wave32).
// GEMMs: V_WMMA_F32_16X16X4_F32, one wave per 16-row panel computing all
// column tiles (A-fragment reuse), K fully unrolled at compile time.
// Aggregation: fp32 global atomics resident in the 192MB L2 (agg = 25.6MB).
// ---------------------------------------------------------------------------

typedef float v2f __attribute__((ext_vector_type(2)));
typedef float v8f __attribute__((ext_vector_type(8)));

#define ELEM_WG 256

// ----------------------------- degree / norm -------------------------------
__global__ void k_init_deg(float* __restrict__ deg, int n) {
  int i = blockIdx.x * blockDim.x + threadIdx.x;
  if (i < n) deg[i] = 1.0f;  // self-loop contribution
}

__global__ void k_deg_edges(const int* __restrict__ dst, float* __restrict__ deg, int e) {
  int i = blockIdx.x * blockDim.x + threadIdx.x;
  if (i < e) atomicAdd(&deg[dst[i]], 1.0f);
}

__global__ void k_rsqrt_inplace(float* __restrict__ d, int n) {
  int i = blockIdx.x * blockDim.x + threadIdx.x;
  if (i < n) d[i] = rsqrtf(d[i]);  // deg >= 1 always (self-loops)
}

// ------------------------------ WMMA GEMM ----------------------------------
// OUT[Nrows x NT*16] = act(X[Nrows x K]) @ W[K x NT*16] (+bias).
// One wave = one 16-row panel; NT accumulators share each A fragment.
// grid = ceil(Nrows/16), block = 32 (one wave, EXEC all ones for WMMA).
//
// Fragment layouts (ISA 7.12.2, 32-bit):
//   A 16x4 : lane l holds row M=l&15, K-pair (l>>4)*2 in 2 VGPRs
//   B 4x16 : lane l holds col N=l&15, K-pair (l>>4)*2 in 2 VGPRs
//   C/D    : VGPR r -> M = r + 8*(l>>4), N = l&15
template <int K, int NT, bool RELU, bool BIAS>
__global__ void k_gemm_wmma(const float* __restrict__ X,
                            const float* __restrict__ W,
                            const float* __restrict__ bias,
                            float* __restrict__ out, int Nrows) {
  constexpr int Hout = NT * 16;
  const int tm   = blockIdx.x;
  const int lane = threadIdx.x;  // 0..31
  const int half = lane >> 4;    // selects K-pair {0,1} vs {2,3}
  const int idx  = lane & 15;

  const int rowA = tm * 16 + idx;
  const int rowS = (rowA < Nrows) ? rowA : (Nrows - 1);  // defensive clamp
  const float* __restrict__ xrow = X + (size_t)rowS * K;

  v8f acc[NT];
#pragma unroll
  for (int t = 0; t < NT; ++t) acc[t] = (v8f){};

#pragma unroll
  for (int kc = 0; kc < K; kc += 4) {
    const int ka = kc + half * 2;                 // even -> 8B aligned
    v2f a = *(const v2f*)(xrow + ka);             // global_load_b64
    if (RELU) { a.x = fmaxf(a.x, 0.0f); a.y = fmaxf(a.y, 0.0f); }
#pragma unroll
    for (int t = 0; t < NT; ++t) {
      const int col = t * 16 + idx;
      v2f b;
      b.x = W[(size_t)ka * Hout + col];
      b.y = W[(size_t)(ka + 1) * Hout + col];
      acc[t] = __builtin_amdgcn_wmma_f32_16x16x4_f32(
          /*neg_a=*/false, a, /*neg_b=*/false, b,
          /*c_mod=*/(short)0, acc[t], /*reuse_a=*/false, /*reuse_b=*/false);
    }
  }

#pragma unroll
  for (int t = 0; t < NT; ++t) {
    const int col = t * 16 + idx;
    const float bv = BIAS ? bias[col] : 0.0f;
#pragma unroll
    for (int r = 0; r < 8; ++r) {
      const int m = tm * 16 + r + half * 8;
      if (m < Nrows) out[(size_t)m * Hout + col] = acc[t][r] + bv;
    }
  }
}

// --------------------------- aggregation (H=64) ----------------------------
// agg[v][f] = dinv[v]^2 * h[v][f] + bias[f]   (self-loop term + bias), float4
__global__ void k_init_agg64(const float4* __restrict__ h,
                             const float* __restrict__ dinv,
                             const float4* __restrict__ bias4,
                             float4* __restrict__ agg, int total4 /* = N*16 */) {
  int i = blockIdx.x * blockDim.x + threadIdx.x;
  if (i >= total4) return;
  const int v = i >> 4;
  const int c = i & 15;
  const float dv = dinv[v];
  const float w  = dv * dv;
  const float4 hv = h[i];
  const float4 bv = bias4[c];
  float4 r;
  r.x = w * hv.x + bv.x;
  r.y = w * hv.y + bv.y;
  r.z = w * hv.z + bv.z;
  r.w = w * hv.w + bv.w;
  agg[i] = r;
}

// agg[dst][f] += dinv[src]*dinv[dst] * h[src][f]; one thread = one edge x 4 feats
__global__ void k_scatter_edges64(const int* __restrict__ src,
                                  const int* __restrict__ dst,
                                  const float* __restrict__ dinv,
                                  const float* __restrict__ h,
                                  float* __restrict__ agg, int e) {
  int t = blockIdx.x * blockDim.x + threadIdx.x;
  if (t >= e * 16) return;
  const int ei = t >> 4;
  const int c  = (t & 15) * 4;
  const int s  = src[ei];
  const int d  = dst[ei];
  const float w = dinv[s] * dinv[d];
  const float4 hv = *(const float4*)(h + (size_t)s * 64 + c);
  float* base = agg + (size_t)d * 64 + c;
  atomicAdd(base + 0, w * hv.x);
  atomicAdd(base + 1, w * hv.y);
  atomicAdd(base + 2, w * hv.z);
  atomicAdd(base + 3, w * hv.w);
}

// ------------------------------- launcher ----------------------------------
extern "C" void kernel_launch(void* const* d_in, const int* in_sizes, int n_in,
                              void* d_out, int out_size, void* d_ws, size_t ws_size,
                              hipStream_t stream) {
  (void)n_in; (void)out_size; (void)ws_size;

  const int F_IN = 128, H = 64;
  const int N = in_sizes[0] / F_IN;   // 100000
  const int E = in_sizes[1] / 2;      // 1200000

  const float* x  = (const float*)d_in[0];
  const int*   ei = (const int*)d_in[1];      // [2,E]: src = ei[0..E), dst = ei[E..2E)
  const float* W1 = (const float*)d_in[2];
  const float* b1 = (const float*)d_in[3];
  const float* W2 = (const float*)d_in[4];
  const float* b2 = (const float*)d_in[5];
  const float* W3 = (const float*)d_in[6];
  const float* b3 = (const float*)d_in[7];
  const float* Wl = (const float*)d_in[8];
  const float* bl = (const float*)d_in[9];
  float* out = (float*)d_out;

  const int* e_src = ei;
  const int* e_dst = ei + E;

  // workspace: dinv[N] | tmp[N*64] | agg[N*64]   (~52 MB)
  float* dinv = (float*)d_ws;
  const size_t dinv_sz = ((size_t)N + 511) & ~(size_t)511;
  float* tmp = dinv + dinv_sz;
  float* agg = tmp + (size_t)N * H;

  const int mt = (N + 15) / 16;  // 6250 row panels (exact)

  // 1) symmetric normalization: dinv = rsqrt(1 + indeg)
  k_init_deg<<<(N + ELEM_WG - 1) / ELEM_WG, ELEM_WG, 0, stream>>>(dinv, N);
  k_deg_edges<<<(E + ELEM_WG - 1) / ELEM_WG, ELEM_WG, 0, stream>>>(e_dst, dinv, E);
  k_rsqrt_inplace<<<(N + ELEM_WG - 1) / ELEM_WG, ELEM_WG, 0, stream>>>(dinv, N);

  const int tot4    = N * 16;                              // N*64 floats / 4
  const int aggGrid = (tot4 + ELEM_WG - 1) / ELEM_WG;
  const int sctGrid = (E * 16 + ELEM_WG - 1) / ELEM_WG;

  // 2) layer 1: tmp = x @ W1 ; agg = scatter(norm * tmp) + b1
  k_gemm_wmma<128, 4, false, false><<<mt, 32, 0, stream>>>(x, W1, nullptr, tmp, N);
  k_init_agg64<<<aggGrid, ELEM_WG, 0, stream>>>(
      (const float4*)tmp, dinv, (const float4*)b1, (float4*)agg, tot4);
  k_scatter_edges64<<<sctGrid, ELEM_WG, 0, stream>>>(e_src, e_dst, dinv, tmp, agg, E);

  // 3) layer 2 (ReLU folded into GEMM A-load)
  k_gemm_wmma<64, 4, true, false><<<mt, 32, 0, stream>>>(agg, W2, nullptr, tmp, N);
  k_init_agg64<<<aggGrid, ELEM_WG, 0, stream>>>(
      (const float4*)tmp, dinv, (const float4*)b2, (float4*)agg, tot4);
  k_scatter_edges64<<<sctGrid, ELEM_WG, 0, stream>>>(e_src, e_dst, dinv, tmp, agg, E);

  // 4) layer 3
  k_gemm_wmma<64, 4, true, false><<<mt, 32, 0, stream>>>(agg, W3, nullptr, tmp, N);
  k_init_agg64<<<aggGrid, ELEM_WG, 0, stream>>>(
      (const float4*)tmp, dinv, (const float4*)b3, (float4*)agg, tot4);
  k_scatter_edges64<<<sctGrid, ELEM_WG, 0, stream>>>(e_src, e_dst, dinv, tmp, agg, E);

  // 5) linear head: out = relu(agg) @ Wl + bl
  k_gemm_wmma<64, 1, true, true><<<mt, 32, 0, stream>>>(agg, Wl, bl, out, N);
}